// BilateralFilter_17076789969463
// MI455X (gfx1250) — compile-verified
//
#include <hip/hip_runtime.h>
#include <hip/hip_bf16.h>
#include <stdint.h>

// Bilateral filter 7x7, fp32, N=8, H=480, W=640.
// gfx1250 strategy: async global->LDS tile staging (GLOBAL_LOAD_ASYNC_TO_LDS_B128
// for the aligned interior, _B32 for halo columns; tracked by ASYNCcnt), then a
// 49-tap LDS stencil using the native v_exp_f32 (exp2) transcendental.

#define KSIZE   7
#define HALO    3
#define TILE_W  64
#define TILE_H  16
#define BLK_Y   4
#define LDS_W   (TILE_W + 2 * HALO)   // 70 valid columns per row
#define LDS_H   (TILE_H + 2 * HALO)   // 22 rows
#define PITCH   72                    // padded row pitch (floats)
#define COLOFF  1                     // column c stored at index c+1 -> interior 16B aligned
#define NTHREADS (TILE_W * BLK_Y)     // 256

// exp(-50*d^2) = exp2(-50*log2(e) * d^2)
#define NEG50_LOG2E (-72.134752044448169364f)

// Per-lane async copies global -> LDS (GVS addressing: SGPR base, 32-bit
// per-lane byte offset in VADDR, LDS byte address in VDST). ASYNCcnt-tracked.
__device__ __forceinline__ void async_copy_b32(unsigned lds_byte_addr,
                                               const float* __restrict__ base,
                                               int byte_off) {
    asm volatile("global_load_async_to_lds_b32 %0, %1, %2"
                 :
                 : "v"(lds_byte_addr), "v"(byte_off), "s"(base)
                 : "memory");
}

__device__ __forceinline__ void async_copy_b128(unsigned lds_byte_addr,
                                                const float* __restrict__ base,
                                                int byte_off) {
    asm volatile("global_load_async_to_lds_b128 %0, %1, %2"
                 :
                 : "v"(lds_byte_addr), "v"(byte_off), "s"(base)
                 : "memory");
}

__device__ __forceinline__ unsigned lds_addr_of(const void* p) {
    // Flat LDS addresses carry the byte offset in the low 32 bits.
    return (unsigned)(uintptr_t)p;
}

__device__ __forceinline__ float fast_exp2(float x) {
#if __has_builtin(__builtin_amdgcn_exp2f)
    return __builtin_amdgcn_exp2f(x);
#else
    return exp2f(x);
#endif
}

__global__ __launch_bounds__(NTHREADS)
void bilateral7x7_kernel(const float* __restrict__ I,
                         const float* __restrict__ g,
                         float* __restrict__ out,
                         int N, int H, int W) {
    __shared__ __align__(16) float tile[LDS_H * PITCH];
    __shared__ float sg[KSIZE * KSIZE];

    const int n   = blockIdx.z;
    const int tx0 = blockIdx.x * TILE_W;
    const int ty0 = blockIdx.y * TILE_H;
    const int tid = threadIdx.y * TILE_W + threadIdx.x;

    const float* baseI = I + (size_t)n * H * W;

    // ---- Stage the 49 spatial Gaussian weights (async b32). ----
    if (tid < KSIZE * KSIZE) {
        async_copy_b32(lds_addr_of(&sg[tid]), g, tid * 4);
    }

    // ---- Stage interior: 22 rows x 16 float4 chunks (16B aligned both sides). ----
    // Chunk k of row r covers global columns tx0+4k..tx0+4k+3, stored at
    // tile[r*PITCH + COLOFF + HALO + 4k]  (byte = 288*r + 16 + 16*k).
    #pragma unroll
    for (int i = tid; i < LDS_H * (TILE_W / 4); i += NTHREADS) {  // 352 tasks
        const int r  = i >> 4;          // / 16 chunks per row
        const int k  = i & 15;
        const int gy = ty0 - HALO + r;
        float* dst = &tile[r * PITCH + COLOFF + HALO + 4 * k];
        if ((unsigned)gy < (unsigned)H) {
            async_copy_b128(lds_addr_of(dst), baseI, (gy * W + tx0 + 4 * k) * 4);
        } else {
            *(float4*)dst = make_float4(0.f, 0.f, 0.f, 0.f);  // zero padding rows
        }
    }

    // ---- Stage halo columns: 22 rows x 6 edge elements (b32). ----
    if (tid < LDS_H * 2 * HALO) {       // 132 tasks
        const int r  = tid / (2 * HALO);
        const int e  = tid - r * (2 * HALO);
        const int c  = (e < HALO) ? e : (TILE_W + e);   // 0,1,2, 67,68,69
        const int gy = ty0 - HALO + r;
        const int gx = tx0 - HALO + c;
        float* dst = &tile[r * PITCH + COLOFF + c];
        if ((unsigned)gy < (unsigned)H && (unsigned)gx < (unsigned)W) {
            async_copy_b32(lds_addr_of(dst), baseI, (gy * W + gx) * 4);
        } else {
            *dst = 0.0f;                // zero padding corners/edges
        }
    }

    // Drain this wave's async copies, then workgroup barrier (also covers the
    // DS zero-stores) so every wave sees the complete tile.
    asm volatile("s_wait_asynccnt 0" ::: "memory");
    __syncthreads();

    const int lx = threadIdx.x;

    #pragma unroll
    for (int py = 0; py < TILE_H / BLK_Y; ++py) {
        const int ly = threadIdx.y + py * BLK_Y;
        const float ctr = tile[(ly + HALO) * PITCH + COLOFF + lx + HALO];
        float num = 0.0f;
        float den = 0.0f;
        #pragma unroll
        for (int dy = 0; dy < KSIZE; ++dy) {
            #pragma unroll
            for (int dx = 0; dx < KSIZE; ++dx) {
                const float v = tile[(ly + dy) * PITCH + COLOFF + lx + dx];
                const float d = v - ctr;
                const float w = fast_exp2(d * d * NEG50_LOG2E) * sg[dy * KSIZE + dx];
                num = fmaf(w, v, num);
                den += w;
            }
        }
        out[((size_t)n * H + (ty0 + ly)) * W + (tx0 + lx)] = __fdividef(num, den);
    }
}

extern "C" void kernel_launch(void* const* d_in, const int* in_sizes, int n_in,
                              void* d_out, int out_size, void* d_ws, size_t ws_size,
                              hipStream_t stream) {
    const float* I = (const float*)d_in[0];   // (8,1,480,640) fp32
    const float* g = (const float*)d_in[1];   // (1,49,1,1) fp32
    float* out = (float*)d_out;               // (8,480,640) fp32

    const int N = 8, H = 480, W = 640;
    dim3 block(TILE_W, BLK_Y, 1);
    dim3 grid(W / TILE_W, H / TILE_H, N);     // 10 x 30 x 8 = 2400 WGs
    bilateral7x7_kernel<<<grid, block, 0, stream>>>(I, g, out, N, H, W);
}